// DynaLoRALinear_91250875171190
// MI455X (gfx1250) — compile-verified
//
#include <hip/hip_runtime.h>
#include <stdint.h>

// Problem constants (from reference)
#define BATCH 8
#define LSEQ  2048
#define DDIM  4096
#define NEXP  4
#define RLORA 8
#define NMODS 7
#define KL    32            // NEXP * RLORA (flattened LoRA rank dim)

typedef __attribute__((ext_vector_type(2))) float v2f;
typedef __attribute__((ext_vector_type(8))) float v8f;
typedef __attribute__((ext_vector_type(4))) unsigned int v4u;
typedef __attribute__((ext_vector_type(8))) int v8i;
typedef __attribute__((ext_vector_type(4))) int v4i;

// Workspace layout (in floats)
#define WS_GATED 0                         // BATCH*DDIM   : pooled @ gating_W.T
#define WS_GATE  (BATCH*DDIM)              // BATCH*NEXP   : gate weights
#define WS_BFLAT (BATCH*DDIM + 64)         // DDIM*KL      : Bflat[n][j] = lora_B[e][n][r]

#if __has_builtin(__builtin_amdgcn_tensor_load_to_lds) && __has_builtin(__builtin_amdgcn_s_wait_tensorcnt)
#define USE_TDM 1
#else
#define USE_TDM 0
#endif

// ---------------------------------------------------------------------------
// Kernel 1: gated[b][n] = dot(x[b, L-1, :], gating_W[n, :])
// ---------------------------------------------------------------------------
__global__ void gated_kernel(const float* __restrict__ x,
                             const float* __restrict__ gW,
                             float* __restrict__ ws) {
  int gid = blockIdx.x * blockDim.x + threadIdx.x;   // 0 .. BATCH*DDIM-1
  int n = gid >> 3;
  int b = gid & 7;
  const float* xr = x + ((size_t)b * LSEQ + (LSEQ - 1)) * DDIM;
  const float* wr = gW + (size_t)n * DDIM;
  float s = 0.f;
  for (int kk = 0; kk < DDIM; kk += 4) {
    float4 xa = *(const float4*)(xr + kk);
    float4 wa = *(const float4*)(wr + kk);
    s += xa.x * wa.x + xa.y * wa.y + xa.z * wa.z + xa.w * wa.w;
  }
  ws[WS_GATED + b * DDIM + n] = s;
}

// ---------------------------------------------------------------------------
// Kernel 2: repack lora_B (E, D, R) -> Bflat (D, KL), Bflat[n][e*R+r]
// ---------------------------------------------------------------------------
__global__ void bflat_kernel(const float* __restrict__ loraB,
                             float* __restrict__ ws) {
  int gid = blockIdx.x * blockDim.x + threadIdx.x;
  if (gid < DDIM * KL) {
    int n = gid >> 5;
    int j = gid & 31;
    int e = j >> 3;
    int r = j & 7;
    ws[WS_BFLAT + gid] = loraB[(size_t)e * DDIM * RLORA + (size_t)n * RLORA + r];
  }
}

// ---------------------------------------------------------------------------
// Kernel 3: logits -> softmax -> module_prob -> rank-based gate weights
// ---------------------------------------------------------------------------
__global__ void gate_kernel(const float* __restrict__ W_r,
                            const float* __restrict__ scores,
                            const int* __restrict__ midx,
                            const int* __restrict__ kptr,
                            float* __restrict__ ws) {
  __shared__ float slog[BATCH][NMODS];
  int t = threadIdx.x;
  if (t < BATCH * NMODS) {
    int b = t / NMODS, m = t % NMODS;
    const float* g = ws + WS_GATED + (size_t)b * DDIM;
    const float* w = W_r + (size_t)m * DDIM;
    float s = 0.f;
    for (int kk = 0; kk < DDIM; kk += 4) {
      float4 ga = *(const float4*)(g + kk);
      float4 wa = *(const float4*)(w + kk);
      s += ga.x * wa.x + ga.y * wa.y + ga.z * wa.z + ga.w * wa.w;
    }
    slog[b][m] = s;
  }
  __syncthreads();
  if (t < BATCH) {
    int b = t;
    float mx = slog[b][0];
    for (int m = 1; m < NMODS; ++m) mx = fmaxf(mx, slog[b][m]);
    float sum = 0.f;
    for (int m = 0; m < NMODS; ++m) sum += __expf(slog[b][m] - mx);
    int   mi = *midx;
    float mp = __expf(slog[b][mi] - mx) / sum;
    int kin = *kptr;
    int klo = kin / 2; if (klo < 1) klo = 1;
    int ke = (mp > 0.5f) ? kin : klo;
    for (int e = 0; e < NEXP; ++e) {
      float se = scores[b * NEXP + e];
      int rank = 0;
      for (int j = 0; j < NEXP; ++j) {          // stable descending rank
        float sj = scores[b * NEXP + j];
        if (sj > se || (sj == se && j < e)) rank++;
      }
      ws[WS_GATE + b * NEXP + e] = (rank < ke) ? 1.f / (float)ke : 0.f;
    }
  }
}

// ---------------------------------------------------------------------------
// Kernel 4: fused out = x @ W_base.T + b_base + (gate .* (x @ A.T)) @ Bflat.T
// Block tile 128M x 128N, 8 waves, wave tile 32M x 64N, FP32 WMMA 16x16x4.
// A/B tiles staged by the Tensor Data Mover (double-buffered, TENSORcnt),
// with D# padding producing the bank-conflict-free LDS stride natively.
// ---------------------------------------------------------------------------
#define BM 128
#define BN 128
#define BK 16
#define LDA 20   // padded LDS row stride (words): 16 data + 4 pad, float4-aligned
#define LDT 36   // padded LDS row stride (words) for 32-wide tiles

#if USE_TDM
// Issue a TDM 2D tile load: rows x 16 fp32, row stride DDIM, LDS pad 4dw / 16dw.
__device__ __forceinline__ void tdm_load_tile(const float* gsrc,
                                              unsigned int lds_byte_off,
                                              int rows) {
  uint64_t ga = (uint64_t)(uintptr_t)gsrc;
  v4u g0;
  g0[0] = 1u;                                            // count=1 (valid user D#)
  g0[1] = lds_byte_off;                                  // lds_addr
  g0[2] = (unsigned int)ga;                              // global_addr[31:0]
  g0[3] = (unsigned int)((ga >> 32) & 0x1FFFFFFu)        // global_addr[56:32]
          | (2u << 30);                                  // type=2 ("image")
  v8i g1;
  g1[0] = (2 << 16)        // data_size = 4 bytes
        | (1 << 20)        // pad_enable
        | (3 << 22)        // pad_interval: 16 DWORDs
        | (3 << 25);       // pad_amount:   4 DWORDs  -> LDS row stride = 20 words
  g1[1] = (int)(((1u << 20) & 0xFFFFu) << 16);           // tensor_dim0 lo16 (=0)
  g1[2] = (int)(((1u << 20) >> 16) |                     // tensor_dim0 hi16
                (((1u << 20) & 0xFFFFu) << 16));         // tensor_dim1 lo16 (=0)
  g1[3] = (int)(((1u << 20) >> 16) | (16u << 16));       // tensor_dim1 hi16 | tile_dim0=16
  g1[4] = rows;                                          // tile_dim1=rows, tile_dim2=0
  g1[5] = DDIM;                                          // tensor_dim0_stride lo32
  g1[6] = 0;                                             // stride0 hi | stride1 lo (unused 2D)
  g1[7] = 0;
  v4i g2;
  g2[0] = (int)(1u << 20);                               // tensor_dim2 (unused, large)
  g2[1] = (int)(1u << 20);                               // tensor_dim3 (unused, large)
  g2[2] = 0;                                             // tensor_dim2_stride lo
  g2[3] = 0;                                             // stride hi | tile_dim3=0
  v4i g3;
  g3[0] = 0;                                             // tensor_dim3_stride lo
  g3[1] = 0;                                             // stride hi | tensor_dim4 lo16 (=0)
  g3[2] = (int)((1u << 20) >> 16);                       // tensor_dim4 hi16 | tile_dim4=0
  g3[3] = 0;
  v8i g4;                                                // extra group (this toolchain's
  g4[0] = 0; g4[1] = 0; g4[2] = 0; g4[3] = 0;            // 6-arg form); zero-filled
  g4[4] = 0; g4[5] = 0; g4[6] = 0; g4[7] = 0;
  __builtin_amdgcn_tensor_load_to_lds(g0, g1, g2, g3, g4, 0);
}
#endif

__global__ __launch_bounds__(256)
void dynalora_main(const float* __restrict__ x,
                   const float* __restrict__ Wb,
                   const float* __restrict__ bb,
                   const float* __restrict__ loraA,
                   const float* __restrict__ ws,
                   float* __restrict__ out) {
  __shared__ float sA [2][BM * LDA];
  __shared__ float sB [2][BN * LDA];
  __shared__ float sAl[2][KL * LDA];
  __shared__ float sT [BM * LDT];
  __shared__ float sBf[BN * LDT];

  const int tid  = threadIdx.x;
  const int lane = tid & 31;
  const int wave = tid >> 5;
  const int wm   = wave >> 1;      // 0..3 : 32-row M strip
  const int wn   = wave & 1;       // 0..1 : 64-col N strip
  const int hl   = lane >> 4;      // half-wave selector
  const int ln   = lane & 15;

  const int mBlock = blockIdx.y * BM;
  const int nBlock = blockIdx.x * BN;
  const int bBatch = mBlock / LSEQ;           // BM divides LSEQ -> one batch per block

  const float* gate  = ws + WS_GATE + bBatch * NEXP;
  const float* Bflat = ws + WS_BFLAT;

  const v8f vzero = {0.f, 0.f, 0.f, 0.f, 0.f, 0.f, 0.f, 0.f};
  v8f acc[2][4];     // 32M x 64N main accumulators
  v8f tac[2][2];     // 32M x 32J LoRA-t accumulators
#pragma unroll
  for (int i = 0; i < 2; ++i) {
#pragma unroll
    for (int j = 0; j < 4; ++j) acc[i][j] = vzero;
#pragma unroll
    for (int j = 0; j < 2; ++j) tac[i][j] = vzero;
  }

  // -------- stage-load helpers (regular path + lora_A tile) --------
  auto load_al = [&](int buf, int kb) {
    if (tid < 128) {
      int row = tid >> 2, c4 = tid & 3;
      *(float4*)&sAl[buf][row * LDA + c4 * 4] =
          *(const float4*)&loraA[((size_t)row) * DDIM + kb + c4 * 4];
    }
  };
#if !USE_TDM
  auto load_ab = [&](int buf, int kb) {
#pragma unroll
    for (int i = 0; i < 2; ++i) {
      int idx = tid + i * 256;
      int row = idx >> 2, c4 = idx & 3;
      *(float4*)&sA[buf][row * LDA + c4 * 4] =
          *(const float4*)&x [((size_t)(mBlock + row)) * DDIM + kb + c4 * 4];
      *(float4*)&sB[buf][row * LDA + c4 * 4] =
          *(const float4*)&Wb[((size_t)(nBlock + row)) * DDIM + kb + c4 * 4];
    }
  };
#endif

  // -------- prologue: stage 0 into buffer 0 --------
#if USE_TDM
  const unsigned int ldsA0 = (unsigned int)(uintptr_t)&sA[0][0];
  const unsigned int ldsA1 = (unsigned int)(uintptr_t)&sA[1][0];
  const unsigned int ldsB0 = (unsigned int)(uintptr_t)&sB[0][0];
  const unsigned int ldsB1 = (unsigned int)(uintptr_t)&sB[1][0];
  if (wave == 0) {
    tdm_load_tile(&x [(size_t)mBlock * DDIM], ldsA0, BM);
    tdm_load_tile(&Wb[(size_t)nBlock * DDIM], ldsB0, BN);
  }
#else
  load_ab(0, 0);
#endif
  load_al(0, 0);

  int buf = 0;
  for (int kb = 0; kb < DDIM; kb += BK) {
#if USE_TDM
    if (wave == 0) __builtin_amdgcn_s_wait_tensorcnt(0);  // stage `buf` DMA done
#endif
    __syncthreads();                       // stage ready; other buffer fully consumed

    const int kn = kb + BK;
    if (kn < DDIM) {
#if USE_TDM
      if (wave == 0) {
        tdm_load_tile(&x [(size_t)mBlock * DDIM + kn], buf ? ldsA0 : ldsA1, BM);
        tdm_load_tile(&Wb[(size_t)nBlock * DDIM + kn], buf ? ldsB0 : ldsB1, BN);
      }
#else
      load_ab(buf ^ 1, kn);
#endif
      load_al(buf ^ 1, kn);
    }

#pragma unroll
    for (int ks = 0; ks < 4; ++ks) {
      const int ko = ks * 4 + hl * 2;
      v2f a[2], bfr[4], al[2];
#pragma unroll
      for (int mi = 0; mi < 2; ++mi)
        a[mi] = *(const v2f*)&sA[buf][(wm * 32 + mi * 16 + ln) * LDA + ko];
#pragma unroll
      for (int ni = 0; ni < 4; ++ni)
        bfr[ni] = *(const v2f*)&sB[buf][(wn * 64 + ni * 16 + ln) * LDA + ko];
#pragma unroll
      for (int ji = 0; ji < 2; ++ji)
        al[ji] = *(const v2f*)&sAl[buf][(ji * 16 + ln) * LDA + ko];
#pragma unroll
      for (int mi = 0; mi < 2; ++mi) {
#pragma unroll
        for (int ni = 0; ni < 4; ++ni)
          acc[mi][ni] = __builtin_amdgcn_wmma_f32_16x16x4_f32(
              false, a[mi], false, bfr[ni], (short)0, acc[mi][ni], false, false);
#pragma unroll
        for (int ji = 0; ji < 2; ++ji)
          tac[mi][ji] = __builtin_amdgcn_wmma_f32_16x16x4_f32(
              false, a[mi], false, al[ji], (short)0, tac[mi][ji], false, false);
      }
    }
    buf ^= 1;
  }

  __syncthreads();
  // Gate-scale t and write to LDS (C layout -> row-major t tile)
#pragma unroll
  for (int mi = 0; mi < 2; ++mi) {
#pragma unroll
    for (int ji = 0; ji < 2; ++ji) {
      int j = ji * 16 + ln;
      float g = gate[j >> 3];
#pragma unroll
      for (int v = 0; v < 8; ++v) {
        int m = wm * 32 + mi * 16 + hl * 8 + v;
        sT[m * LDT + j] = tac[mi][ji][v] * g;
      }
    }
  }
  // Stage Bflat tile: 128 rows x 32 cols = 1024 float4 -> 4 per thread
#pragma unroll
  for (int i = 0; i < 4; ++i) {
    int idx = tid + i * 256;
    int row = idx >> 3, c4 = idx & 7;
    *(float4*)&sBf[row * LDT + c4 * 4] =
        *(const float4*)&Bflat[(size_t)(nBlock + row) * KL + c4 * 4];
  }
  __syncthreads();

  // Rank-32 LoRA GEMM: acc += t_scaled @ Bflat.T
#pragma unroll
  for (int ks = 0; ks < 8; ++ks) {
    const int ko = ks * 4 + hl * 2;
    v2f a2[2], b2[4];
#pragma unroll
    for (int mi = 0; mi < 2; ++mi)
      a2[mi] = *(const v2f*)&sT[(wm * 32 + mi * 16 + ln) * LDT + ko];
#pragma unroll
    for (int ni = 0; ni < 4; ++ni)
      b2[ni] = *(const v2f*)&sBf[(wn * 64 + ni * 16 + ln) * LDT + ko];
#pragma unroll
    for (int mi = 0; mi < 2; ++mi)
#pragma unroll
      for (int ni = 0; ni < 4; ++ni)
        acc[mi][ni] = __builtin_amdgcn_wmma_f32_16x16x4_f32(
            false, a2[mi], false, b2[ni], (short)0, acc[mi][ni], false, false);
  }

  // Epilogue: bias + store
#pragma unroll
  for (int ni = 0; ni < 4; ++ni) {
    int n = nBlock + wn * 64 + ni * 16 + ln;
    float bias = bb[n];
#pragma unroll
    for (int mi = 0; mi < 2; ++mi) {
#pragma unroll
      for (int v = 0; v < 8; ++v) {
        int m = mBlock + wm * 32 + mi * 16 + hl * 8 + v;
        out[(size_t)m * DDIM + n] = acc[mi][ni][v] + bias;
      }
    }
  }
}

// ---------------------------------------------------------------------------
extern "C" void kernel_launch(void* const* d_in, const int* in_sizes, int n_in,
                              void* d_out, int out_size, void* d_ws, size_t ws_size,
                              hipStream_t stream) {
  const float* x      = (const float*)d_in[0];
  const float* scores = (const float*)d_in[1];
  const float* Wb     = (const float*)d_in[2];
  const float* bb     = (const float*)d_in[3];
  const float* gW     = (const float*)d_in[4];
  const float* Wr     = (const float*)d_in[5];
  const float* lA     = (const float*)d_in[6];
  const float* lB     = (const float*)d_in[7];
  const int*   midx   = (const int*)d_in[8];
  const int*   kptr   = (const int*)d_in[9];
  float* out = (float*)d_out;
  float* ws  = (float*)d_ws;

  gated_kernel<<<dim3((BATCH * DDIM) / 256), 256, 0, stream>>>(x, gW, ws);
  bflat_kernel<<<dim3((DDIM * KL + 255) / 256), 256, 0, stream>>>(lB, ws);
  gate_kernel<<<dim3(1), 256, 0, stream>>>(Wr, scores, midx, kptr, ws);

  dim3 grid(DDIM / BN, (BATCH * LSEQ) / BM);   // N-fast order -> x M-tile L2 reuse
  dynalora_main<<<grid, 256, 0, stream>>>(x, Wb, bb, lA, ws, out);
}